// ProductVectorQuantize_57913339020074
// MI455X (gfx1250) — compile-verified
//
#include <hip/hip_runtime.h>
#include <stdint.h>
#include <math.h>

// Problem constants (from reference)
#define B_   16
#define H_   64
#define WT_  1200
#define C_   64
#define G_   8
#define K_   8
#define S_   1024
#define T_   300          // Wt / OVERLAP

static constexpr int ZQN    = B_ * H_ * WT_ * C_;   // 78,643,200 floats (z_q_out)
static constexpr int CODESN = B_ * G_ * T_;         // 38,400 (codes, stored as float)
static constexpr int LOSS0  = ZQN + CODESN;         // cb_loss slot, cm_loss = +1
static constexpr int EN_WS  = G_ * S_ * K_;         // 65,536 floats of normalized embed in ws
static constexpr float INV_CNT = 1.0f / (float)(B_ * T_ * G_ * K_);  // 1/307200

typedef __attribute__((ext_vector_type(2))) float        v2f;
typedef __attribute__((ext_vector_type(8))) float        v8f;
typedef __attribute__((ext_vector_type(4))) unsigned int v4u;
typedef __attribute__((ext_vector_type(8))) int          v8s;
typedef __attribute__((ext_vector_type(4))) int          v4s;

__device__ __forceinline__ v8f wmma4(v2f a, v2f b, v8f c) {
  // V_WMMA_F32_16X16X4_F32 : D(16x16,f32) = A(16x4,f32) * B(4x16,f32) + C
  return __builtin_amdgcn_wmma_f32_16x16x4_f32(false, a, false, b, (short)0, c, false, false);
}

// --------------------- TDM issue helpers (raw D# words) ---------------------
__device__ __forceinline__ void tdm_load_raw(v4u g0, v8s g1, v4s g2, v4s g3) {
#if defined(__clang_major__) && __clang_major__ >= 23
  v8s z8 = {0, 0, 0, 0, 0, 0, 0, 0};
  __builtin_amdgcn_tensor_load_to_lds(g0, g1, g2, g3, z8, 0);
#else
  __builtin_amdgcn_tensor_load_to_lds(g0, g1, g2, g3, 0);
#endif
}
__device__ __forceinline__ void tdm_store_raw(v4u g0, v8s g1, v4s g2, v4s g3) {
#if defined(__clang_major__) && __clang_major__ >= 23
  v8s z8 = {0, 0, 0, 0, 0, 0, 0, 0};
  __builtin_amdgcn_tensor_store_from_lds(g0, g1, g2, g3, z8, 0);
#else
  __builtin_amdgcn_tensor_store_from_lds(g0, g1, g2, g3, 0);
#endif
}

// Group0: count=1, lds_addr, 57b global byte address, type=2 ("image")
__device__ __forceinline__ v4u tdm_g0(const void* gaddr, unsigned ldsAddr) {
  unsigned long long ga = (unsigned long long)(uintptr_t)gaddr;
  v4u g0;
  g0.x = 1u;                                   // count=1, user descriptor
  g0.y = ldsAddr;                              // LDS byte address
  g0.z = (unsigned)ga;                         // global_addr[31:0]
  g0.w = (unsigned)(ga >> 32) | (2u << 30);    // global_addr[56:32] | type=2
  return g0;
}

// Phase-1 load: 2D tile (x = 4096 f32 contiguous, y = 4 h-slices, stride 76800
// elements).  LDS padding: +4 DWORDs every 256 DWORDs -> 260-float rows
// (bank-conflict-free for the WMMA A-fragment reads).  OOB x (tail tiles)
// zero-fills.
__device__ __forceinline__ void tdm_load_tile(const float* gsrc, unsigned ldsAddr,
                                              unsigned dim0) {
  v8s g1;
  g1[0] = (2 << 16) | (1 << 20) | (7 << 22) | (3 << 25); // 4B elems, pad 4dw/256dw
  g1[1] = (int)((dim0 & 0xffffu) << 16);                 // tensor_dim0 lo16
  g1[2] = (int)((dim0 >> 16) | (4u << 16));              // tensor_dim0 hi | tensor_dim1=4
  g1[3] = (int)(4096u << 16);                            // tile_dim0=4096
  g1[4] = 4;                                             // tile_dim1=4, tile_dim2=0
  g1[5] = 76800;                                         // tensor_dim0_stride lo32
  g1[6] = 0;
  g1[7] = 0;
  v4s g2; g2[0] = 1; g2[1] = 1; g2[2] = 0; g2[3] = 0;    // inert dims 2/3
  v4s g3; g3[0] = 0; g3[1] = (1 << 16); g3[2] = 0; g3[3] = 0; // tensor_dim4=1
  tdm_load_raw(tdm_g0(gsrc, ldsAddr), g1, g2, g3);
}

// Phase-3 store: 4D tile. x=c (32 f32, 128B line), y=m (16, stride 256),
// z=dh (4, stride 76800), zz=g-pair (4, stride 64).  tensor_dim1=validM drops
// tail-tile rows in hardware.
__device__ __forceinline__ void tdm_store_tile(float* gdst, unsigned ldsAddr,
                                               unsigned validM) {
  v8s g1;
  g1[0] = (2 << 16);                                     // 4B elems, no pad
  g1[1] = (int)(32u << 16);                              // tensor_dim0 = 32
  g1[2] = (int)(validM << 16);                           // tensor_dim1 = validM
  g1[3] = (32 << 16);                                    // tile_dim0 = 32
  g1[4] = 16 | (4 << 16);                                // tile_dim1=16, tile_dim2=4
  g1[5] = 256;                                           // stride0 = 256
  g1[6] = (int)((76800u & 0xffffu) << 16);               // stride1 lo16 (76800)
  g1[7] = (int)(76800u >> 16);                           // stride1 hi32
  v4s g2;
  g2[0] = 4;                                             // tensor_dim2 = 4
  g2[1] = 4;                                             // tensor_dim3 = 4
  g2[2] = 64;                                            // tensor_dim2_stride = 64
  g2[3] = (4 << 16);                                     // tile_dim3 = 4
  v4s g3; g3[0] = 0; g3[1] = (1 << 16); g3[2] = 0; g3[3] = 0; // tensor_dim4=1
  tdm_store_raw(tdm_g0(gdst, ldsAddr), g1, g2, g3);
}

// ---------------------------------------------------------------------------
// Prep: normalize embed rows into ws[0..65535]; zero the loss accumulator.
// ---------------------------------------------------------------------------
__global__ void pvq_prep(const float* __restrict__ embed, float* __restrict__ ws) {
  int i = blockIdx.x * 256 + threadIdx.x;           // 0..8191 = (g,s)
  if (i == 0) ws[EN_WS] = 0.0f;                     // loss accumulator
  if (i < G_ * S_) {
    float v[8];
    float s = 0.0f;
#pragma unroll
    for (int k = 0; k < 8; ++k) { v[k] = embed[i * 8 + k]; s += v[k] * v[k]; }
    float inv = 1.0f / fmaxf(sqrtf(s), 1e-12f);     // matches clip(norm,1e-12)
#pragma unroll
    for (int k = 0; k < 8; ++k) ws[i * 8 + k] = v[k] * inv;
  }
}

// ---------------------------------------------------------------------------
// Main: one workgroup per (b, 16-t tile). 256 threads = 8 waves; wave == group.
// Dynamic LDS: bufA[16640] | bufB[16640] (TDM double buffers, also phase-3
// store staging) | zd_s[1024] | zq_s[1024]  = 141,312 B.
// ---------------------------------------------------------------------------
__global__ __launch_bounds__(256) void pvq_main(
    const float* __restrict__ ze, const float* __restrict__ Wd,
    const float* __restrict__ Wu, const float* __restrict__ enw,
    float* __restrict__ out, float* __restrict__ lossAcc)
{
  extern __shared__ __align__(16) float smem[];
  float* bufA = smem;            // 16640 floats: 64 rows x (256 + 4 pad)
  float* bufB = smem + 16640;
  float* zd_s = smem + 33280;    // 1024: zd[g][m][k]
  float* zq_s = smem + 34304;    // 1024: zq[g][m][k]

  const int tid  = threadIdx.x;
  const int g    = tid >> 5;     // wave id = product group
  const int lane = tid & 31;
  const int n    = lane & 15;    // WMMA column / row-in-tile index
  const int half = lane >> 4;    // K-half selector in fragments
  const int t0   = blockIdx.x * 16;
  const int b    = blockIdx.y;

  const unsigned ldsA = (unsigned)(uintptr_t)(void*)bufA;  // low 32 bits = LDS offset
  const unsigned ldsB = (unsigned)(uintptr_t)(void*)bufB;
  const int rem       = (WT_ - 4 * t0) * 64;               // valid x extent (elements)
  const unsigned dim0 = rem < 4096 ? (unsigned)rem : 4096u;

  // ======================= Phase 1: zd = Z * W_down^T ======================
  if (g == 0)
    tdm_load_tile(ze + ((size_t)(b * 64) * WT_ + 4 * t0) * 64, ldsA, dim0);

  v8f acc = {};
  const float bmask  = (n < 8) ? 1.0f : 0.0f;              // only K=8 cols valid
  const int   wdbase = (g * 8 + (n & 7)) * 2048 + half * 2;
  for (int hc = 0; hc < 16; ++hc) {
    if (hc + 1 < 16) {
      if (g == 0)
        tdm_load_tile(ze + ((size_t)(b * 64 + (hc + 1) * 4) * WT_ + 4 * t0) * 64,
                      ((hc + 1) & 1) ? ldsB : ldsA, dim0);
      __builtin_amdgcn_s_wait_tensorcnt(1);   // chunk hc landed; hc+1 streaming
    } else {
      __builtin_amdgcn_s_wait_tensorcnt(0);
    }
    __syncthreads();
    const float* zb    = (hc & 1) ? bufB : bufA;
    const int    abase = (half * 2 * 16 + n) * 260 + (g >> 1) * 64 + ((g & 1) << 5);
    for (int cc = 0; cc < 32; ++cc) {
      v2f a;
      a.x = zb[abase + cc];                   // kk = 2*half
      a.y = zb[abase + cc + 16 * 260];        // kk = 2*half + 1
      const float2 wv = *(const float2*)(Wd + (wdbase + cc * 64 + hc * 4));
      v2f bf; bf.x = wv.x * bmask; bf.y = wv.y * bmask;
      acc = wmma4(a, bf, acc);
    }
    __syncthreads();                          // reads done before buffer reuse
  }

  // Transpose zd fragment (cols across lanes) -> zd_s[g][m][k]
  if (n < 8) {
#pragma unroll
    for (int i = 0; i < 8; ++i)
      zd_s[(g * 16 + half * 8 + i) * 8 + n] = acc[i];
  }
  __syncthreads();

  // ================== Phase 2: argmax over codebook, zq, loss ==============
  const int m = n;                                  // row of the t-tile
  float zdv[8];
#pragma unroll
  for (int k = 0; k < 8; ++k) zdv[k] = zd_s[(g * 16 + m) * 8 + k];

  float best = -3.4e38f;
  int bi = 0;
  const float* eg = enw + g * (S_ * 8);
  for (int j = 0; j < 512; ++j) {                   // lanes 0-15: even s, 16-31: odd s
    int s = 2 * j + half;
    const float4 e0 = *(const float4*)(eg + s * 8);
    const float4 e1 = *(const float4*)(eg + s * 8 + 4);
    float d = zdv[0] * e0.x + zdv[1] * e0.y + zdv[2] * e0.z + zdv[3] * e0.w +
              zdv[4] * e1.x + zdv[5] * e1.y + zdv[6] * e1.z + zdv[7] * e1.w;
    if (d > best) { best = d; bi = s; }
  }
  {
    float ob = __shfl_xor(best, 16);
    int   oi = __shfl_xor(bi, 16);
    if (ob > best || (ob == best && oi < bi)) { best = ob; bi = oi; }
  }
  const int t = t0 + m;
  float lsum = 0.0f;
  if (half == 0) {
    if (t < T_) out[ZQN + (b * 8 + g) * T_ + t] = (float)bi;   // codes (B,G,T)
    const float* e = eg + bi * 8;
#pragma unroll
    for (int k = 0; k < 8; ++k) {
      float q = e[k];
      zq_s[(g * 16 + m) * 8 + k] = q;
      float df = q - zdv[k];
      lsum += df * df;
    }
    if (t >= T_) lsum = 0.0f;
  }
#pragma unroll
  for (int off = 16; off > 0; off >>= 1) lsum += __shfl_xor(lsum, off);
  if (lane == 0) atomicAdd(lossAcc, lsum);
  __syncthreads();

  // ======================= Phase 3: zu = zq * W_up^T =======================
  v2f a0, a1;
  {
    const float* zq = zq_s + (g * 16 + m) * 8;
    int kb = half * 2;
    a0.x = zq[kb];     a0.y = zq[kb + 1];           // K 0..3
    a1.x = zq[kb + 4]; a1.y = zq[kb + 5];           // K 4..7
  }
  const unsigned vm = (unsigned)((T_ - t0) < 16 ? (T_ - t0) : 16);
  for (int hc = 0; hc < 16; ++hc) {
    __builtin_amdgcn_s_wait_tensorcnt(2);           // chunk hc-2 stores retired
    __syncthreads();                                // staging buffer free for all waves
    float* sb = (hc & 1) ? bufB : bufA;             // parity p half at sb + p*8192
    for (int dh = 0; dh < 4; ++dh) {
      for (int ct = 0; ct < 2; ++ct) {
        int d = (n + 16 * ct) * 64 + hc * 4 + dh;
        const float* wp = Wu + (g * 2048 + d) * 8 + half * 2;
        float2 w0 = *(const float2*)(wp);
        float2 w1 = *(const float2*)(wp + 4);
        v2f b0; b0.x = w0.x; b0.y = w0.y;
        v2f b1; b1.x = w1.x; b1.y = w1.y;
        v8f od = {};
        od = wmma4(a0, b0, od);
        od = wmma4(a1, b1, od);
        int cl = n + 16 * ct;                       // c_local 0..31
        float* dst = sb + (g & 1) * 8192 +
                     ((((g >> 1) * 4 + dh) * 16) + half * 8) * 32 + cl;
#pragma unroll
        for (int i = 0; i < 8; ++i) dst[i * 32] = od[i];
      }
    }
    __syncthreads();                                // staging complete
    if (g == 0) {
      float*   gbase = out + ((size_t)(b * 64 + hc * 4) * WT_ + 4 * t0) * 64;
      unsigned lbase = (hc & 1) ? ldsB : ldsA;
      tdm_store_tile(gbase,      lbase,              vm);  // groups 0,2,4,6 (c 0..31)
      tdm_store_tile(gbase + 32, lbase + 8192 * 4,   vm);  // groups 1,3,5,7 (c 32..63)
    }
  }
  // outstanding TDM stores drained by S_ENDPGM's implicit wait-idle
}

// ---------------------------------------------------------------------------
// Finish: scale the loss sum; cb_loss == cm_loss numerically.
// ---------------------------------------------------------------------------
__global__ void pvq_fin(const float* __restrict__ ws, float* __restrict__ out) {
  if (threadIdx.x == 0 && blockIdx.x == 0) {
    float mloss = ws[EN_WS] * INV_CNT;
    out[LOSS0]     = mloss;
    out[LOSS0 + 1] = mloss;
  }
}

extern "C" void kernel_launch(void* const* d_in, const int* in_sizes, int n_in,
                              void* d_out, int out_size, void* d_ws, size_t ws_size,
                              hipStream_t stream) {
  (void)in_sizes; (void)n_in; (void)out_size; (void)ws_size;
  const float* ze    = (const float*)d_in[0];   // z_e    (16, 76800, 64)
  const float* Wd    = (const float*)d_in[1];   // W_down (8, 8, 2048)
  const float* Wu    = (const float*)d_in[2];   // W_up   (8, 2048, 8)
  const float* embed = (const float*)d_in[3];   // embed  (8, 1024, 8)
  float* out = (float*)d_out;
  float* ws  = (float*)d_ws;                    // [0..65535]=en_norm, [65536]=loss acc

  pvq_prep<<<32, 256, 0, stream>>>(embed, ws);

  dim3 grid((T_ + 15) / 16, B_);                // 19 x 16
  size_t shmem = (16640 * 2 + 1024 + 1024) * sizeof(float);  // 141,312 B
  pvq_main<<<grid, 256, shmem, stream>>>(ze, Wd, Wu, ws, out, ws + EN_WS);

  pvq_fin<<<1, 64, 0, stream>>>(ws, out);
}